// AttentionPool_1297080123655
// MI455X (gfx1250) — compile-verified
//
#include <hip/hip_runtime.h>
#include <math.h>

typedef __attribute__((ext_vector_type(2))) float v2f;
typedef __attribute__((ext_vector_type(8))) float v8f;

#define IN_CH     128
#define OUT_CH    64
#define NHEAD     4
#define TOT_OUT   256   // NHEAD*OUT_CH
#define NEG_SLOPE 0.2f

#define MTILE 64        // nodes per block tile
#define KCH   32        // K-chunk staged in LDS
#define KPAD  34        // padded LDS row stride (conflict-free 16-lane frag loads)

// ---------------------------------------------------------------- init
__global__ void k_init(float* out, float* denom, unsigned* segmax, int C) {
    int i = blockIdx.x * blockDim.x + threadIdx.x;
    if (i < C * TOT_OUT) out[i] = 0.0f;
    if (i < 4 * C) { denom[i] = 0.0f; segmax[i] = 0xFF800000u; } // -inf bits
}

// ------------------------------------------------- fold att_w into lin_w
// v[h][i] = sum_d lin_w[h*64+d][i] * att_w[d] ; c[h] = b_h . att_w + att_b
__global__ void k_fold(const float* __restrict__ lin_w, const float* __restrict__ lin_b,
                       const float* __restrict__ att_w, const float* __restrict__ att_b,
                       float* v, float* c) {
    int t = blockIdx.x * blockDim.x + threadIdx.x;
    if (t < NHEAD * IN_CH) {
        int h = t / IN_CH, i = t % IN_CH;
        float s = 0.0f;
        for (int d = 0; d < OUT_CH; ++d)
            s += lin_w[(size_t)(h * OUT_CH + d) * IN_CH + i] * att_w[d];
        v[h * IN_CH + i] = s;
    }
    if (t < NHEAD) {
        float s = att_b[0];
        for (int d = 0; d < OUT_CH; ++d) s += lin_b[t * OUT_CH + d] * att_w[d];
        c[t] = s;
    }
}

// ------------------------------------------ score + leakyrelu + seg-max
__global__ __launch_bounds__(256)
void k_score(const float* __restrict__ x, const int* __restrict__ y,
             const float* __restrict__ v, const float* __restrict__ c,
             float* __restrict__ score, int* segmax_i, unsigned* segmax_u, int Nn) {
    __shared__ float vl[NHEAD * IN_CH];
    __shared__ float cl[NHEAD];
    int t = threadIdx.x;
    vl[t] = v[t]; vl[t + 256] = v[t + 256];
    if (t < NHEAD) cl[t] = c[t];
    __syncthreads();

    int node = blockIdx.x * 256 + t;
    if (node >= Nn) return;

    const float4* xp = (const float4*)(x + (size_t)node * IN_CH);
    float s[NHEAD] = {0.f, 0.f, 0.f, 0.f};
#pragma unroll
    for (int q = 0; q < IN_CH / 4; ++q) {
        float4 xv = xp[q];
#pragma unroll
        for (int h = 0; h < NHEAD; ++h) {
            const float* vh = &vl[h * IN_CH + q * 4];
            s[h] += xv.x * vh[0] + xv.y * vh[1] + xv.z * vh[2] + xv.w * vh[3];
        }
    }
    int yy = y[node];
#pragma unroll
    for (int h = 0; h < NHEAD; ++h) {
        float r = s[h] + cl[h];
        r = (r >= 0.0f) ? r : NEG_SLOPE * r;           // leaky_relu(0.2)
        score[(size_t)node * NHEAD + h] = r;
        int idx = yy * NHEAD + h;
        // monotone float max via sign-split integer atomics (no CAS loop)
        if (r >= 0.0f) atomicMax(&segmax_i[idx], __float_as_int(r));
        else           atomicMin(&segmax_u[idx], (unsigned)__float_as_int(r));
    }
}

// -------------------------------------------------- exp-sum per segment
__global__ void k_denom(const float* __restrict__ score, const int* __restrict__ y,
                        const float* __restrict__ segmax, float* denom, int Nn) {
    int i = blockIdx.x * blockDim.x + threadIdx.x;
    if (i >= 4 * Nn) return;
    int node = i >> 2, h = i & 3;
    int idx = y[node] * NHEAD + h;
    atomicAdd(&denom[idx], __expf(score[i] - segmax[idx]));
}

// ---------------------- WMMA GEMM + weighted segment scatter (main pass)
__global__ __launch_bounds__(256)
void k_gemm(const float* __restrict__ x, const int* __restrict__ y,
            const float* __restrict__ lin_w, const float* __restrict__ lin_b,
            const float* __restrict__ score, const float* __restrict__ segmax,
            const float* __restrict__ denom, float* __restrict__ out, int Nn) {
    __shared__ float Xl[MTILE * KPAD];      //  8.7 KB
    __shared__ float Wl[TOT_OUT * KPAD];    // 34.8 KB  (n-major)
    __shared__ float wgt[MTILE * NHEAD];    // per-node per-head softmax weight
    __shared__ float bl[TOT_OUT];
    __shared__ int   yl[MTILE];

    int t = threadIdx.x;
    int blockM = blockIdx.x * MTILE;

    // ---- preamble: softmax weights, segment ids, bias into LDS
    {
        int m = t >> 2, h = t & 3;
        int node = blockM + m;
        float w = 0.0f; int yy = 0;
        if (node < Nn) {
            yy = y[node];
            int idx = yy * NHEAD + h;
            w = __expf(score[(size_t)node * NHEAD + h] - segmax[idx]) / denom[idx];
        }
        wgt[m * NHEAD + h] = w;
        if (h == 0) yl[m] = yy;
        bl[t] = lin_b[t];
    }

    int wave = t >> 5, lane = t & 31;
    int rowbase = (wave >> 1) * 16;          // 4 row-groups of 16
    int colbase = (wave & 1) * 128;          // 2 col-groups of 128
    int lm      = lane & 15;
    int khalf   = (lane < 16) ? 0 : 2;       // A/B frag: lanes 16-31 hold K+2,K+3

    v8f acc[8];
    const v8f vzero = {0.f, 0.f, 0.f, 0.f, 0.f, 0.f, 0.f, 0.f};
#pragma unroll
    for (int i = 0; i < 8; ++i) acc[i] = vzero;

    for (int kc = 0; kc < IN_CH / KCH; ++kc) {
        int kbase = kc * KCH;
        __syncthreads();
        // ---- stage X chunk [MTILE x KCH]
        {
            int row = t >> 2, q = t & 3;
            int node = blockM + row;
#pragma unroll
            for (int j = 0; j < 2; ++j) {
                int kk = q * 8 + j * 4;
                float4 xv = make_float4(0.f, 0.f, 0.f, 0.f);
                if (node < Nn)
                    xv = *(const float4*)(x + (size_t)node * IN_CH + kbase + kk);
                Xl[row * KPAD + kk + 0] = xv.x;
                Xl[row * KPAD + kk + 1] = xv.y;
                Xl[row * KPAD + kk + 2] = xv.z;
                Xl[row * KPAD + kk + 3] = xv.w;
            }
            if (kc + 1 < IN_CH / KCH && node < Nn)
                __builtin_prefetch(x + (size_t)node * IN_CH + kbase + KCH, 0, 1);
        }
        // ---- stage W chunk [TOT_OUT x KCH], n-major
        {
            int n = t;
#pragma unroll
            for (int q = 0; q < 8; ++q) {
                int kk = q * 4;
                float4 w4 = *(const float4*)(lin_w + (size_t)n * IN_CH + kbase + kk);
                Wl[n * KPAD + kk + 0] = w4.x;
                Wl[n * KPAD + kk + 1] = w4.y;
                Wl[n * KPAD + kk + 2] = w4.z;
                Wl[n * KPAD + kk + 3] = w4.w;
            }
            if (kc + 1 < IN_CH / KCH)
                __builtin_prefetch(lin_w + (size_t)n * IN_CH + kbase + KCH, 0, 1);
        }
        __syncthreads();
        // ---- compute: 8 k-steps of 4, 8 N-tiles of 16
#pragma unroll
        for (int ks = 0; ks < KCH / 4; ++ks) {
            int kk = ks * 4 + khalf;
            v2f a = *(const v2f*)(&Xl[(rowbase + lm) * KPAD + kk]);
#pragma unroll
            for (int nt = 0; nt < 8; ++nt) {
                int n = colbase + nt * 16 + lm;
                v2f b = *(const v2f*)(&Wl[n * KPAD + kk]);
                acc[nt] = __builtin_amdgcn_wmma_f32_16x16x4_f32(
                    false, a, false, b, (short)0, acc[nt], false, false);
            }
        }
    }

    // ---- epilogue: (acc + bias) * weight, scatter into pooled[y]
    int mhalf = (lane < 16) ? 0 : 8;
#pragma unroll
    for (int nt = 0; nt < 8; ++nt) {
        int n = colbase + nt * 16 + lm;
        int head = n >> 6;
        float bn = bl[n];
#pragma unroll
        for (int r = 0; r < 8; ++r) {
            int m = rowbase + mhalf + r;
            float w = wgt[m * NHEAD + head];
            float val = (acc[nt][r] + bn) * w;     // w==0 for OOB rows -> adds 0
            atomicAdd(&out[(size_t)yl[m] * TOT_OUT + n], val);
        }
    }
}

// ---------------------------------------------------------------- launch
extern "C" void kernel_launch(void* const* d_in, const int* in_sizes, int n_in,
                              void* d_out, int out_size, void* d_ws, size_t ws_size,
                              hipStream_t stream) {
    (void)n_in; (void)ws_size;
    const float* x     = (const float*)d_in[0];
    const int*   y     = (const int*)  d_in[1];
    const float* lin_w = (const float*)d_in[3];
    const float* lin_b = (const float*)d_in[4];
    const float* att_w = (const float*)d_in[5];
    const float* att_b = (const float*)d_in[6];
    float* out = (float*)d_out;

    int N = in_sizes[0] / IN_CH;
    int C = out_size / TOT_OUT;

    float* wsf    = (float*)d_ws;
    float* v      = wsf;                 // 512
    float* c      = wsf + 512;           // 4
    float* segmax = wsf + 516;           // 4C
    float* denom  = segmax + 4 * C;      // 4C
    float* score  = denom + 4 * C;       // 4N

    int initN = C * TOT_OUT;
    k_init <<<(initN + 255) / 256, 256, 0, stream>>>(out, denom, (unsigned*)segmax, C);
    k_fold <<<2, 256, 0, stream>>>(lin_w, lin_b, att_w, att_b, v, c);
    k_score<<<(N + 255) / 256, 256, 0, stream>>>(x, y, v, c, score,
                                                 (int*)segmax, (unsigned*)segmax, N);
    k_denom<<<(4 * N + 255) / 256, 256, 0, stream>>>(score, y, segmax, denom, N);
    k_gemm <<<(N + MTILE - 1) / MTILE, 256, 0, stream>>>(x, y, lin_w, lin_b,
                                                         score, segmax, denom, out, N);
}